// DecoderRNN_764504178845
// MI455X (gfx1250) — compile-verified
//
#include <hip/hip_runtime.h>
#include <math.h>

#define HIDDEN 1024
#define VOCAB  50257

typedef __attribute__((ext_vector_type(2))) float v2f;
typedef __attribute__((ext_vector_type(8))) float v8f;

// One wave computes y[n0..n0+15] = W[n0..n0+15, :] . x (+ bias) with
// V_WMMA_F32_16X16X4_F32, accumulating over K=HIDDEN in steps of 4.
//   A (16x4, MxK): lane m (0..15) holds W[n0+m, kc+0], W[n0+m, kc+1] in (a0,a1);
//                  lane m+16 holds W[n0+m, kc+2], W[n0+m, kc+3].     (K split (0,2)/(1,3))
//   B (4x16, KxN): x[kc+k] broadcast across all N columns -> every column of D
//                  ends up equal to the desired dot product.
//   D (16x16):     lane 0 acc[i] = y[n0+i], lane 16 acc[i] = y[n0+8+i].
__device__ __forceinline__ void wave_matvec16(
    const float* __restrict__ W,      // row-major [rowsTotal x HIDDEN]
    const float* xs,                  // LDS-resident x[HIDDEN]
    int n0, int rowsTotal,
    const float* __restrict__ bias,
    float* __restrict__ yout)
{
  const int lane = threadIdx.x & 31;
  int row = n0 + (lane & 15);
  if (row >= rowsTotal) row = rowsTotal - 1;          // clamp: EXEC stays all-ones
  const int koff = (lane >> 4) << 1;                  // 0 for lanes 0-15, 2 for 16-31
  const float* pA = W + (size_t)row * HIDDEN + koff;  // streams 16B/row per 4-K step
  const float* pB = xs + koff;                        // LDS broadcast per half-wave

  v8f acc = {};
#pragma unroll 4
  for (int kc = 0; kc < HIDDEN; kc += 4) {
    v2f a = __builtin_nontemporal_load((const v2f*)(pA + kc));  // weights: read-once, NT
    v2f b = *(const v2f*)(pB + kc);                             // ds_load_b64 broadcast
    acc = __builtin_amdgcn_wmma_f32_16x16x4_f32(
        /*neg_a=*/false, a, /*neg_b=*/false, b,
        /*c_mod=*/(short)0, acc, /*reuse_a=*/false, /*reuse_b=*/false);
  }

  if ((lane & 15) == 0) {                             // lanes 0 and 16 hold column N=0
    const int base = n0 + ((lane >> 4) << 3);
#pragma unroll
    for (int i = 0; i < 8; ++i) {
      const int r = base + i;
      if (r < rowsTotal) yout[r] = acc[i] + bias[r];
    }
  }
}

// Kernel 1: gi = W_ih . relu(emb[tok]) + b_ih   (blocks [0,48))
//           gh = W_hh . h               + b_hh   (blocks [48,96))
__global__ void __launch_bounds__(128) k_gru_matvec(
    const int* __restrict__ tok, const float* __restrict__ hidden,
    const float* __restrict__ emb,
    const float* __restrict__ w_ih, const float* __restrict__ w_hh,
    const float* __restrict__ b_ih, const float* __restrict__ b_hh,
    float* __restrict__ gi, float* __restrict__ gh)
{
  __shared__ float xs[HIDDEN];
  const int tilesPerMat = (3 * HIDDEN) / 64;          // 48 blocks of 64 rows
  const bool isIH = (int)blockIdx.x < tilesPerMat;

  if (isIH) {
    const float* er = emb + (size_t)tok[0] * HIDDEN;
    for (int i = threadIdx.x; i < HIDDEN; i += 128) {
      float v = er[i];
      xs[i] = v > 0.f ? v : 0.f;                      // fused relu(embedding)
    }
  } else {
    for (int i = threadIdx.x; i < HIDDEN; i += 128) xs[i] = hidden[i];
  }
  __syncthreads();

  const int tile = isIH ? blockIdx.x : blockIdx.x - tilesPerMat;
  const int n0 = tile * 64 + (threadIdx.x >> 5) * 16;
  wave_matvec16(isIH ? w_ih : w_hh, xs, n0, 3 * HIDDEN,
                isIH ? b_ih : b_hh, isIH ? gi : gh);
}

// Kernel 2: GRU gate math (PyTorch order r,z,n) -> h_new
__global__ void k_gru_combine(
    const float* __restrict__ hidden, const float* __restrict__ gi,
    const float* __restrict__ gh, float* __restrict__ h_new,
    float* __restrict__ out_h)
{
  const int i = blockIdx.x * blockDim.x + threadIdx.x;
  if (i >= HIDDEN) return;
  const float r = 1.f / (1.f + expf(-(gi[i] + gh[i])));
  const float z = 1.f / (1.f + expf(-(gi[HIDDEN + i] + gh[HIDDEN + i])));
  const float n = tanhf(gi[2 * HIDDEN + i] + r * gh[2 * HIDDEN + i]);
  const float hv = (1.f - z) * n + z * hidden[i];
  h_new[i] = hv;
  out_h[i] = hv;                                      // second tuple output
}

// Kernel 3: logits = W_out . h_new + b_out   (dominant 206 MB stream)
__global__ void __launch_bounds__(128) k_logits_matvec(
    const float* __restrict__ h_new, const float* __restrict__ w_out,
    const float* __restrict__ b_out, float* __restrict__ logits)
{
  __shared__ float xs[HIDDEN];
  for (int i = threadIdx.x; i < HIDDEN; i += 128) xs[i] = h_new[i];
  __syncthreads();
  const int n0 = blockIdx.x * 64 + (threadIdx.x >> 5) * 16;
  wave_matvec16(w_out, xs, n0, VOCAB, b_out, logits);
}

// Kernel 4: logZ = max + log(sum exp(logits - max)), single workgroup (L2-hot)
__global__ void __launch_bounds__(1024) k_softmax_reduce(
    const float* __restrict__ logits, float* __restrict__ logZ)
{
  __shared__ float red[1024];
  const int t = threadIdx.x;
  float m = -INFINITY;
  for (int i = t; i < VOCAB; i += 1024) m = fmaxf(m, logits[i]);
  red[t] = m; __syncthreads();
  for (int s = 512; s > 0; s >>= 1) {
    if (t < s) red[t] = fmaxf(red[t], red[t + s]);
    __syncthreads();
  }
  const float mx = red[0];
  __syncthreads();
  float sum = 0.f;
  for (int i = t; i < VOCAB; i += 1024) sum += expf(logits[i] - mx);
  red[t] = sum; __syncthreads();
  for (int s = 512; s > 0; s >>= 1) {
    if (t < s) red[t] += red[t + s];
    __syncthreads();
  }
  if (t == 0) *logZ = mx + logf(red[0]);
}

// Kernel 5: logp = logits - logZ
__global__ void k_logp_write(
    const float* __restrict__ logits, const float* __restrict__ logZ,
    float* __restrict__ out)
{
  const int i = blockIdx.x * blockDim.x + threadIdx.x;
  if (i < VOCAB) out[i] = logits[i] - logZ[0];
}

extern "C" void kernel_launch(void* const* d_in, const int* in_sizes, int n_in,
                              void* d_out, int out_size, void* d_ws, size_t ws_size,
                              hipStream_t stream) {
  const int*   tok    = (const int*)d_in[0];
  const float* hidden = (const float*)d_in[1];
  const float* emb    = (const float*)d_in[2];
  const float* w_ih   = (const float*)d_in[3];
  const float* w_hh   = (const float*)d_in[4];
  const float* b_ih   = (const float*)d_in[5];
  const float* b_hh   = (const float*)d_in[6];
  const float* w_out  = (const float*)d_in[7];
  const float* b_out  = (const float*)d_in[8];
  float* out = (float*)d_out;           // [VOCAB logp][HIDDEN h_new]
  float* ws  = (float*)d_ws;

  float* gi     = ws;                   // 3072
  float* gh     = ws + 3 * HIDDEN;      // 3072
  float* h_new  = ws + 6 * HIDDEN;      // 1024
  float* logits = ws + 7 * HIDDEN;      // 50257
  float* logZ   = logits + VOCAB;       // 1      (~230 KB total)

  k_gru_matvec<<<96, 128, 0, stream>>>(tok, hidden, emb, w_ih, w_hh,
                                       b_ih, b_hh, gi, gh);
  k_gru_combine<<<(HIDDEN + 127) / 128, 128, 0, stream>>>(hidden, gi, gh,
                                                          h_new, out + VOCAB);
  k_logits_matvec<<<(VOCAB + 63) / 64, 128, 0, stream>>>(h_new, w_out, b_out,
                                                         logits);
  k_softmax_reduce<<<1, 1024, 0, stream>>>(logits, logZ);
  k_logp_write<<<(VOCAB + 255) / 256, 256, 0, stream>>>(logits, logZ, out);
}